// HNM_35304631173972
// MI455X (gfx1250) — compile-verified
//
#include <hip/hip_runtime.h>
#include <cstdint>

typedef float v2f __attribute__((ext_vector_type(2)));
typedef float v8f __attribute__((ext_vector_type(8)));

#define NROWS 4096
#define FEATR 891
#define FEATP 896
#define TOPK  455
#define SIGMA_INV 0.1f
#define BN_EPS 1e-5f

__device__ __forceinline__ v8f wmma_f32(v2f a, v2f b, v8f c) {
  // V_WMMA_F32_16X16X4_F32: D = A(16x4) x B(4x16) + C(16x16), fp32
  return __builtin_amdgcn_wmma_f32_16x16x4_f32(false, a, false, b, (short)0, c,
                                               false, false);
}

// ---------------------------------------------------------------------------
// K0: fused conv1/conv2/conv3 (+ReLU) per sample -> feat[4096][896] (zero-pad
// cols 891..895) and sq[n] = ||feat_n||^2.  One block (128 thr) per sample.
// ---------------------------------------------------------------------------
__global__ __launch_bounds__(128) void conv_feat_kernel(
    const float* __restrict__ x,
    const float* __restrict__ w1c, const float* __restrict__ b1c,
    const float* __restrict__ w2c, const float* __restrict__ b2c,
    const float* __restrict__ w3c, const float* __restrict__ b3c,
    float* __restrict__ feat, float* __restrict__ sq) {
  __shared__ float sx[103 * 25];        // input sample
  __shared__ float sh1[2 * 101 * 9];    // h1 with D-padding rows
  __shared__ float sh2[4 * 101];        // h2 with D-padding
  __shared__ float sw1[126]; __shared__ float sb1[2];
  __shared__ float sw2[216]; __shared__ float sb2[4];
  __shared__ float sw3[108]; __shared__ float sb3[9];
  __shared__ float red[128];
  const int tid = threadIdx.x;
  const int n = blockIdx.x;
  const float* xs = x + (size_t)n * 2575;
  for (int i = tid; i < 2575; i += 128) sx[i] = xs[i];
  for (int i = tid; i < 126; i += 128) sw1[i] = w1c[i];
  for (int i = tid; i < 216; i += 128) sw2[i] = w2c[i];
  for (int i = tid; i < 108; i += 128) sw3[i] = w3c[i];
  if (tid < 2) sb1[tid] = b1c[tid];
  if (tid < 4) sb2[tid] = b2c[tid];
  if (tid < 9) sb3[tid] = b3c[tid];
  for (int i = tid; i < 2 * 101 * 9; i += 128) sh1[i] = 0.f;
  for (int i = tid; i < 4 * 101; i += 128) sh2[i] = 0.f;
  __syncthreads();
  // conv1: [1,103,5,5] -> [2,99,3,3], kernel (7,3,3), pad D=1
  for (int idx = tid; idx < 1782; idx += 128) {
    int co = idx / 891, rem = idx % 891;
    int d = rem / 9, hw = rem % 9, i0 = hw / 3, j0 = hw % 3;
    float acc = sb1[co];
    for (int kd = 0; kd < 7; ++kd) {
      int dd = d - 1 + kd;
      if (dd < 0 || dd > 102) continue;
      const float* xp = &sx[dd * 25 + i0 * 5 + j0];
      const float* wp = &sw1[co * 63 + kd * 9];
#pragma unroll
      for (int ki = 0; ki < 3; ++ki)
#pragma unroll
        for (int kj = 0; kj < 3; ++kj) acc += xp[ki * 5 + kj] * wp[ki * 3 + kj];
    }
    sh1[co * 909 + (d + 1) * 9 + hw] = fmaxf(acc, 0.f);
  }
  __syncthreads();
  // conv2: [2,99,3,3] -> [4,99,1,1], kernel (3,3,3), pad D=1
  for (int idx = tid; idx < 396; idx += 128) {
    int co = idx / 99, d = idx % 99;
    float acc = sb2[co];
    for (int ci = 0; ci < 2; ++ci)
      for (int kd = 0; kd < 3; ++kd) {
        const float* hp = &sh1[ci * 909 + (d + kd) * 9];
        const float* wp = &sw2[co * 54 + ci * 27 + kd * 9];
#pragma unroll
        for (int hw = 0; hw < 9; ++hw) acc += hp[hw] * wp[hw];
      }
    sh2[co * 101 + d + 1] = fmaxf(acc, 0.f);
  }
  __syncthreads();
  // conv3: [4,99] -> [9,99], kernel (3,1,1), pad D=1;  feat + sumsq
  float ssq = 0.f;
  for (int idx = tid; idx < FEATR; idx += 128) {
    int co = idx / 99, d = idx % 99;
    float acc = sb3[co];
#pragma unroll
    for (int ci = 0; ci < 4; ++ci)
#pragma unroll
      for (int kd = 0; kd < 3; ++kd)
        acc += sh2[ci * 101 + d + kd] * sw3[co * 12 + ci * 3 + kd];
    float v = fmaxf(acc, 0.f);
    feat[(size_t)n * FEATP + idx] = v;
    ssq += v * v;
  }
  if (tid < FEATP - FEATR) feat[(size_t)n * FEATP + FEATR + tid] = 0.f;
  red[tid] = ssq;
  __syncthreads();
  for (int s = 64; s > 0; s >>= 1) {
    if (tid < s) red[tid] += red[tid + s];
    __syncthreads();
  }
  if (tid == 0) sq[n] = red[0];
}

// ---------------------------------------------------------------------------
// K1: a_e = exp(-max(sq_i + sq_j - 2 * feat@feat^T, 0)/10)  via fp32 WMMA.
// 128x128 tile per block, 8 waves, each wave = 32x64 (2x4 WMMA tiles).
// ---------------------------------------------------------------------------
__global__ __launch_bounds__(256) void gram_kernel(
    const float* __restrict__ feat, const float* __restrict__ sq,
    float* __restrict__ ae) {
  __shared__ float At[128 * 18];
  __shared__ float Bt[128 * 18];
  __shared__ float sqR[128];
  __shared__ float sqC[128];
  const int tid = threadIdx.x;
  const int rowBase = blockIdx.x * 128;
  const int colBase = blockIdx.y * 128;
  if (tid < 128) sqR[tid] = sq[rowBase + tid];
  else           sqC[tid - 128] = sq[colBase + tid - 128];
  const int wave = tid >> 5, lane = tid & 31;
  const int wr = wave >> 1, wc = wave & 1;
  const int lhi = lane >> 4, llo = lane & 15;
  v8f acc[2][4] = {};
  const int r = tid >> 1, h = tid & 1;
  const float* gA = feat + (size_t)(rowBase + r) * FEATP + h * 8;
  const float* gB = feat + (size_t)(colBase + r) * FEATP + h * 8;
  float* lA = &At[r * 18 + h * 8];
  float* lB = &Bt[r * 18 + h * 8];
  for (int k0 = 0; k0 < FEATP; k0 += 16) {
    float4 a0 = *(const float4*)(gA + k0);
    float4 a1 = *(const float4*)(gA + k0 + 4);
    float4 b0 = *(const float4*)(gB + k0);
    float4 b1 = *(const float4*)(gB + k0 + 4);
    __syncthreads();
    ((float2*)lA)[0] = make_float2(a0.x, a0.y);
    ((float2*)lA)[1] = make_float2(a0.z, a0.w);
    ((float2*)lA)[2] = make_float2(a1.x, a1.y);
    ((float2*)lA)[3] = make_float2(a1.z, a1.w);
    ((float2*)lB)[0] = make_float2(b0.x, b0.y);
    ((float2*)lB)[1] = make_float2(b0.z, b0.w);
    ((float2*)lB)[2] = make_float2(b1.x, b1.y);
    ((float2*)lB)[3] = make_float2(b1.z, b1.w);
    __syncthreads();
#pragma unroll
    for (int kk = 0; kk < 16; kk += 4) {
      const int off = kk + 2 * lhi;
      v2f aF[2], bF[4];
#pragma unroll
      for (int rt = 0; rt < 2; ++rt)
        aF[rt] = *(const v2f*)&At[(wr * 32 + rt * 16 + llo) * 18 + off];
#pragma unroll
      for (int ct = 0; ct < 4; ++ct)
        bF[ct] = *(const v2f*)&Bt[(wc * 64 + ct * 16 + llo) * 18 + off];
#pragma unroll
      for (int rt = 0; rt < 2; ++rt)
#pragma unroll
        for (int ct = 0; ct < 4; ++ct)
          acc[rt][ct] = wmma_f32(aF[rt], bF[ct], acc[rt][ct]);
    }
  }
#pragma unroll
  for (int rt = 0; rt < 2; ++rt)
#pragma unroll
    for (int ct = 0; ct < 4; ++ct) {
      const float* av = (const float*)&acc[rt][ct];
#pragma unroll
      for (int v = 0; v < 8; ++v) {
        int m = v + lhi * 8;
        int il = wr * 32 + rt * 16 + m;
        int jl = wc * 64 + ct * 16 + llo;
        float d2 = fmaxf(sqR[il] + sqC[jl] - 2.f * av[v], 0.f);
        ae[(size_t)(rowBase + il) * NROWS + (colBase + jl)] =
            expf(-d2 * SIGMA_INV);
      }
    }
}

// ---------------------------------------------------------------------------
// K2: per-row exact 455-th largest via bitwise radix select on float bits
// (a_e >= 0 so uint order == float order), then D^-1/2.
// ---------------------------------------------------------------------------
__global__ __launch_bounds__(256) void rowsel_kernel(
    const float* __restrict__ ae, float* __restrict__ th,
    float* __restrict__ dinv) {
  __shared__ int scnt;
  __shared__ float ssum;
  const int i = blockIdx.x, tid = threadIdx.x;
  const float* row = ae + (size_t)i * NROWS;
  float v[16];
  unsigned u[16];
#pragma unroll
  for (int t = 0; t < 16; ++t) {
    v[t] = row[tid + t * 256];
    u[t] = __float_as_uint(v[t]);
  }
  unsigned prefix = 0u;
  for (int bit = 30; bit >= 0; --bit) {   // sign bit always 0
    unsigned cand = prefix | (1u << bit);
    int c = 0;
#pragma unroll
    for (int t = 0; t < 16; ++t) c += (u[t] >= cand) ? 1 : 0;
    if (tid == 0) scnt = 0;
    __syncthreads();
    atomicAdd(&scnt, c);
    __syncthreads();
    int total = scnt;
    __syncthreads();
    if (total >= TOPK) prefix = cand;
  }
  float thv = __uint_as_float(prefix);
  // d_i = sum_j A_ij + 1 ; A_ii = 1 + 0.5*ae_ii, A_ij = 0.9*ae_ij (topk)
  float s = 0.f;
#pragma unroll
  for (int t = 0; t < 16; ++t) {
    int j = tid + t * 256;
    float val = v[t];
    if (j == i)            s += 1.0f + 0.5f * val;
    else if (val >= thv)   s += 0.9f * val;
  }
  if (tid == 0) ssum = 0.f;
  __syncthreads();
  atomicAdd(&ssum, s);
  __syncthreads();
  if (tid == 0) {
    th[i] = thv;
    dinv[i] = 1.f / sqrtf(ssum + 1.f);
  }
}

// ---------------------------------------------------------------------------
// K4: Y1[j][c] = dinv[j] * (feat @ w1)[j][c]   (M=4096,K=891->896,N=128) WMMA
// ---------------------------------------------------------------------------
__global__ __launch_bounds__(256) void xw1_kernel(
    const float* __restrict__ feat, const float* __restrict__ w1,
    const float* __restrict__ dinv, float* __restrict__ Y1) {
  __shared__ float At[128 * 18];
  __shared__ float Bt[128 * 18];
  __shared__ float dR[128];
  const int tid = threadIdx.x;
  const int rowBase = blockIdx.x * 128;
  if (tid < 128) dR[tid] = dinv[rowBase + tid];
  const int wave = tid >> 5, lane = tid & 31;
  const int wr = wave >> 1, wc = wave & 1;
  const int lhi = lane >> 4, llo = lane & 15;
  v8f acc[2][4] = {};
  const int r = tid >> 1, h = tid & 1;
  const float* gA = feat + (size_t)(rowBase + r) * FEATP + h * 8;
  float* lA = &At[r * 18 + h * 8];
  const int bn = tid & 127, bh = tid >> 7;
  for (int k0 = 0; k0 < FEATP; k0 += 16) {
    float4 a0 = *(const float4*)(gA + k0);
    float4 a1 = *(const float4*)(gA + k0 + 4);
    float bv[8];
#pragma unroll
    for (int it = 0; it < 8; ++it) {
      int kg = k0 + bh * 8 + it;
      bv[it] = (kg < FEATR) ? w1[(size_t)kg * 128 + bn] : 0.f;
    }
    __syncthreads();
    ((float2*)lA)[0] = make_float2(a0.x, a0.y);
    ((float2*)lA)[1] = make_float2(a0.z, a0.w);
    ((float2*)lA)[2] = make_float2(a1.x, a1.y);
    ((float2*)lA)[3] = make_float2(a1.z, a1.w);
#pragma unroll
    for (int it = 0; it < 8; ++it) Bt[bn * 18 + bh * 8 + it] = bv[it];
    __syncthreads();
#pragma unroll
    for (int kk = 0; kk < 16; kk += 4) {
      const int off = kk + 2 * lhi;
      v2f aF[2], bF[4];
#pragma unroll
      for (int rt = 0; rt < 2; ++rt)
        aF[rt] = *(const v2f*)&At[(wr * 32 + rt * 16 + llo) * 18 + off];
#pragma unroll
      for (int ct = 0; ct < 4; ++ct)
        bF[ct] = *(const v2f*)&Bt[(wc * 64 + ct * 16 + llo) * 18 + off];
#pragma unroll
      for (int rt = 0; rt < 2; ++rt)
#pragma unroll
        for (int ct = 0; ct < 4; ++ct)
          acc[rt][ct] = wmma_f32(aF[rt], bF[ct], acc[rt][ct]);
    }
  }
#pragma unroll
  for (int rt = 0; rt < 2; ++rt)
#pragma unroll
    for (int ct = 0; ct < 4; ++ct) {
      const float* av = (const float*)&acc[rt][ct];
#pragma unroll
      for (int v = 0; v < 8; ++v) {
        int m = v + lhi * 8;
        int il = wr * 32 + rt * 16 + m;
        int jn = wc * 64 + ct * 16 + llo;
        Y1[(size_t)(rowBase + il) * 128 + jn] = dR[il] * av[v];
      }
    }
}

// ---------------------------------------------------------------------------
// K5: G1[i][c] = dinv[i] * sum_j A_ij * Y1[j][c]  (A built on the fly from
// a_e + per-row threshold; K=4096, N=128) WMMA
// ---------------------------------------------------------------------------
__global__ __launch_bounds__(256) void g1_kernel(
    const float* __restrict__ ae, const float* __restrict__ th,
    const float* __restrict__ dinv, const float* __restrict__ Y1,
    float* __restrict__ G1) {
  __shared__ float At[128 * 18];
  __shared__ float Bt[128 * 18];
  __shared__ float dR[128];
  const int tid = threadIdx.x;
  const int rowBase = blockIdx.x * 128;
  if (tid < 128) dR[tid] = dinv[rowBase + tid];
  const int wave = tid >> 5, lane = tid & 31;
  const int wr = wave >> 1, wc = wave & 1;
  const int lhi = lane >> 4, llo = lane & 15;
  v8f acc[2][4] = {};
  const int r = tid >> 1, h = tid & 1;
  const int gi = rowBase + r;
  const float thr = th[gi];
  const float* gArow = ae + (size_t)gi * NROWS + h * 8;
  float* lA = &At[r * 18 + h * 8];
  const int bn = tid & 127, bh = tid >> 7;
  for (int k0 = 0; k0 < NROWS; k0 += 16) {
    float4 a0 = *(const float4*)(gArow + k0);
    float4 a1 = *(const float4*)(gArow + k0 + 4);
    float av8[8] = {a0.x, a0.y, a0.z, a0.w, a1.x, a1.y, a1.z, a1.w};
    float tv[8];
#pragma unroll
    for (int e = 0; e < 8; ++e) {
      int j = k0 + h * 8 + e;
      float val = av8[e];
      tv[e] = (j == gi) ? (1.0f + 0.5f * val)
                        : ((val >= thr) ? 0.9f * val : 0.f);
    }
    float bv[8];
#pragma unroll
    for (int it = 0; it < 8; ++it)
      bv[it] = Y1[(size_t)(k0 + bh * 8 + it) * 128 + bn];
    __syncthreads();
#pragma unroll
    for (int e = 0; e < 8; ++e) lA[e] = tv[e];
#pragma unroll
    for (int it = 0; it < 8; ++it) Bt[bn * 18 + bh * 8 + it] = bv[it];
    __syncthreads();
#pragma unroll
    for (int kk = 0; kk < 16; kk += 4) {
      const int off = kk + 2 * lhi;
      v2f aF[2], bF[4];
#pragma unroll
      for (int rt = 0; rt < 2; ++rt)
        aF[rt] = *(const v2f*)&At[(wr * 32 + rt * 16 + llo) * 18 + off];
#pragma unroll
      for (int ct = 0; ct < 4; ++ct)
        bF[ct] = *(const v2f*)&Bt[(wc * 64 + ct * 16 + llo) * 18 + off];
#pragma unroll
      for (int rt = 0; rt < 2; ++rt)
#pragma unroll
        for (int ct = 0; ct < 4; ++ct)
          acc[rt][ct] = wmma_f32(aF[rt], bF[ct], acc[rt][ct]);
    }
  }
#pragma unroll
  for (int rt = 0; rt < 2; ++rt)
#pragma unroll
    for (int ct = 0; ct < 4; ++ct) {
      const float* av = (const float*)&acc[rt][ct];
#pragma unroll
      for (int v = 0; v < 8; ++v) {
        int m = v + lhi * 8;
        int il = wr * 32 + rt * 16 + m;
        int jn = wc * 64 + ct * 16 + llo;
        G1[(size_t)(rowBase + il) * 128 + jn] = dR[il] * av[v];
      }
    }
}

// ---------------------------------------------------------------------------
// K8: G2[i][c] = dinv[i] * sum_j A_ij * Y2[j][c]  (N=16, 1 WMMA tile/wave)
// ---------------------------------------------------------------------------
__global__ __launch_bounds__(256) void g2_kernel(
    const float* __restrict__ ae, const float* __restrict__ th,
    const float* __restrict__ dinv, const float* __restrict__ Y2,
    float* __restrict__ G2) {
  __shared__ float At[128 * 18];
  __shared__ float Bt[16 * 18];
  __shared__ float dR[128];
  const int tid = threadIdx.x;
  const int rowBase = blockIdx.x * 128;
  if (tid < 128) dR[tid] = dinv[rowBase + tid];
  const int wave = tid >> 5, lane = tid & 31;
  const int lhi = lane >> 4, llo = lane & 15;
  v8f acc = {};
  const int r = tid >> 1, h = tid & 1;
  const int gi = rowBase + r;
  const float thr = th[gi];
  const float* gArow = ae + (size_t)gi * NROWS + h * 8;
  float* lA = &At[r * 18 + h * 8];
  const int bkk = tid >> 4, bnn = tid & 15;
  for (int k0 = 0; k0 < NROWS; k0 += 16) {
    float4 a0 = *(const float4*)(gArow + k0);
    float4 a1 = *(const float4*)(gArow + k0 + 4);
    float av8[8] = {a0.x, a0.y, a0.z, a0.w, a1.x, a1.y, a1.z, a1.w};
    float tv[8];
#pragma unroll
    for (int e = 0; e < 8; ++e) {
      int j = k0 + h * 8 + e;
      float val = av8[e];
      tv[e] = (j == gi) ? (1.0f + 0.5f * val)
                        : ((val >= thr) ? 0.9f * val : 0.f);
    }
    float bval = Y2[(size_t)(k0 + bkk) * 16 + bnn];
    __syncthreads();
#pragma unroll
    for (int e = 0; e < 8; ++e) lA[e] = tv[e];
    Bt[bnn * 18 + bkk] = bval;
    __syncthreads();
#pragma unroll
    for (int kk = 0; kk < 16; kk += 4) {
      const int off = kk + 2 * lhi;
      v2f aF = *(const v2f*)&At[(wave * 16 + llo) * 18 + off];
      v2f bF = *(const v2f*)&Bt[llo * 18 + off];
      acc = wmma_f32(aF, bF, acc);
    }
  }
  const float* av = (const float*)&acc;
#pragma unroll
  for (int v = 0; v < 8; ++v) {
    int m = v + lhi * 8;
    int il = wave * 16 + m;
    G2[(size_t)(rowBase + il) * 16 + llo] = dR[il] * av[v];
  }
}

// ---------------------------------------------------------------------------
// BatchNorm stats per column over 4096 rows (biased variance)
// ---------------------------------------------------------------------------
__global__ __launch_bounds__(256) void bnstats_kernel(
    const float* __restrict__ X, int ld, float* __restrict__ mean,
    float* __restrict__ rstd) {
  __shared__ float r1[256];
  __shared__ float r2[256];
  const int c = blockIdx.x, tid = threadIdx.x;
  float s = 0.f, ss = 0.f;
#pragma unroll
  for (int t = 0; t < 16; ++t) {
    float v = X[(size_t)(tid + t * 256) * ld + c];
    s += v;
    ss += v * v;
  }
  r1[tid] = s;
  r2[tid] = ss;
  __syncthreads();
  for (int st = 128; st > 0; st >>= 1) {
    if (tid < st) {
      r1[tid] += r1[tid + st];
      r2[tid] += r2[tid + st];
    }
    __syncthreads();
  }
  if (tid == 0) {
    float m = r1[0] * (1.f / 4096.f);
    float var = r2[0] * (1.f / 4096.f) - m * m;
    mean[c] = m;
    rstd[c] = 1.f / sqrtf(var + BN_EPS);
  }
}

// ---------------------------------------------------------------------------
// K7: g1 = tanh(bn1(G1)); Y2[j][0:9] = dinv[j] * (g1 @ w2)[j]; pad to 16
// ---------------------------------------------------------------------------
__global__ __launch_bounds__(256) void g1w2_kernel(
    const float* __restrict__ G1, const float* __restrict__ mean,
    const float* __restrict__ rstd, const float* __restrict__ gamma,
    const float* __restrict__ beta, const float* __restrict__ w2,
    const float* __restrict__ dinv, float* __restrict__ Y2) {
  __shared__ float sm[128], sr[128], sg[128], sb[128], sw[128 * 9];
  const int tid = threadIdx.x;
  if (tid < 128) {
    sm[tid] = mean[tid];
    sr[tid] = rstd[tid];
    sg[tid] = gamma[tid];
    sb[tid] = beta[tid];
  }
  for (int i = tid; i < 128 * 9; i += 256) sw[i] = w2[i];
  __syncthreads();
  const int j = blockIdx.x * 256 + tid;
  float acc[9] = {};
  for (int c = 0; c < 128; ++c) {
    float g = tanhf(sg[c] * (G1[(size_t)j * 128 + c] - sm[c]) * sr[c] + sb[c]);
#pragma unroll
    for (int k = 0; k < 9; ++k) acc[k] += g * sw[c * 9 + k];
  }
  float dj = dinv[j];
#pragma unroll
  for (int k = 0; k < 9; ++k) Y2[(size_t)j * 16 + k] = dj * acc[k];
#pragma unroll
  for (int k = 9; k < 16; ++k) Y2[(size_t)j * 16 + k] = 0.f;
}

// ---------------------------------------------------------------------------
// K10: out = tanh(bn2(G2))  -> [4096,9]
// ---------------------------------------------------------------------------
__global__ __launch_bounds__(256) void out_kernel(
    const float* __restrict__ G2, const float* __restrict__ mean,
    const float* __restrict__ rstd, const float* __restrict__ gamma,
    const float* __restrict__ beta, float* __restrict__ out) {
  int idx = blockIdx.x * 256 + threadIdx.x;
  if (idx >= NROWS * 9) return;
  int j = idx / 9, c = idx % 9;
  float g = G2[(size_t)j * 16 + c];
  out[idx] = tanhf(gamma[c] * (g - mean[c]) * rstd[c] + beta[c]);
}

// ---------------------------------------------------------------------------
extern "C" void kernel_launch(void* const* d_in, const int* in_sizes, int n_in,
                              void* d_out, int out_size, void* d_ws,
                              size_t ws_size, hipStream_t stream) {
  (void)in_sizes; (void)n_in; (void)out_size; (void)ws_size;
  const float* x    = (const float*)d_in[0];
  const float* c1w  = (const float*)d_in[1];
  const float* c1b  = (const float*)d_in[2];
  const float* c2w  = (const float*)d_in[3];
  const float* c2b  = (const float*)d_in[4];
  const float* c3w  = (const float*)d_in[5];
  const float* c3b  = (const float*)d_in[6];
  const float* w1   = (const float*)d_in[7];
  const float* w2   = (const float*)d_in[8];
  const float* bn1g = (const float*)d_in[9];
  const float* bn1b = (const float*)d_in[10];
  const float* bn2g = (const float*)d_in[11];
  const float* bn2b = (const float*)d_in[12];

  float* ws = (float*)d_ws;
  size_t o = 0;
  float* feat = ws + o; o += (size_t)NROWS * FEATP;
  float* sq   = ws + o; o += NROWS;
  float* thv  = ws + o; o += NROWS;
  float* dinv = ws + o; o += NROWS;
  float* ae   = ws + o; o += (size_t)NROWS * NROWS;
  float* Y1   = ws + o; o += (size_t)NROWS * 128;
  float* G1   = ws + o; o += (size_t)NROWS * 128;
  float* Y2   = ws + o; o += (size_t)NROWS * 16;
  float* G2   = ws + o; o += (size_t)NROWS * 16;
  float* m1   = ws + o; o += 128;
  float* r1   = ws + o; o += 128;
  float* m2   = ws + o; o += 16;
  float* r2   = ws + o; o += 16;

  conv_feat_kernel<<<NROWS, 128, 0, stream>>>(x, c1w, c1b, c2w, c2b, c3w, c3b,
                                              feat, sq);
  dim3 gg(32, 32);
  gram_kernel<<<gg, 256, 0, stream>>>(feat, sq, ae);
  rowsel_kernel<<<NROWS, 256, 0, stream>>>(ae, thv, dinv);
  xw1_kernel<<<32, 256, 0, stream>>>(feat, w1, dinv, Y1);
  g1_kernel<<<32, 256, 0, stream>>>(ae, thv, dinv, Y1, G1);
  bnstats_kernel<<<128, 256, 0, stream>>>(G1, 128, m1, r1);
  g1w2_kernel<<<16, 256, 0, stream>>>(G1, m1, r1, bn1g, bn1b, w2, dinv, Y2);
  g2_kernel<<<32, 256, 0, stream>>>(ae, thv, dinv, Y2, G2);
  bnstats_kernel<<<9, 256, 0, stream>>>(G2, 16, m2, r2);
  out_kernel<<<(NROWS * 9 + 255) / 256, 256, 0, stream>>>(G2, m2, r2, bn2g,
                                                          bn2b, (float*)d_out);
}